// RNNDecoder_51445118272104
// MI455X (gfx1250) — compile-verified
//
#include <hip/hip_runtime.h>
#include <math.h>

typedef __attribute__((ext_vector_type(2))) float v2f;
typedef __attribute__((ext_vector_type(8))) float v8f;

#define HID   512
#define EMBED 512
#define BSZ   128
#define SRC_L 128
#define VOCAB 50257

#define KC    64          // K-chunk staged in LDS
#define KCP   68          // padded LDS row stride (68 % 64banks = 4 -> conflict-free)

// ---------------------------------------------------------------------------
// rnn_in = [embedding[x] | context_vec]   -> [128, 1024]
// ---------------------------------------------------------------------------
__global__ void build_rnn_in_kernel(const int* __restrict__ x,
                                    const float* __restrict__ embedding,
                                    const float* __restrict__ context,
                                    float* __restrict__ rnn_in) {
  int idx = blockIdx.x * blockDim.x + threadIdx.x;  // 128*1024 threads
  int b = idx >> 10;
  int j = idx & 1023;
  float v;
  if (j < EMBED) v = embedding[(size_t)x[b] * EMBED + j];
  else           v = context[b * HID + (j - EMBED)];
  rnn_in[idx] = v;
}

// ---------------------------------------------------------------------------
// M=128 WMMA fp32 GEMM:  C[128,N] = A[128,K] @ W[N,K]^T (+bias) (opt tanh)
//
// Block = 8 waves; each wave owns one 16-wide N tile, all 8 M tiles.
// A is staged chunk-by-chunk in LDS (shared by all 8 waves -> A read once
// per block instead of once per wave). W is streamed from HBM with
// register-batched loads (8 k-steps per clause) so the dominant fc1 weight
// stream stays latency-hidden. K must be a multiple of 64.
// ---------------------------------------------------------------------------
__global__ void __launch_bounds__(256)
gemm128_wmma(const float* __restrict__ A,
             const float* __restrict__ W,
             const float* __restrict__ bias,
             float* __restrict__ C,
             int N, int K, int ldc, int fuse_tanh) {
  __shared__ float a_lds[128 * KCP];                  // 34.8 KB

  const int tid    = threadIdx.y * 32 + threadIdx.x;  // 0..255
  const int lane   = threadIdx.x;                     // 0..31
  const int ntile  = blockIdx.x * 8 + threadIdx.y;
  const int ntiles = (N + 15) >> 4;
  const bool active = (ntile < ntiles);
  const int n0   = (active ? ntile : 0) << 4;
  const int half = lane >> 4;                         // 0 or 1
  const int l    = lane & 15;

  // B fragment source: W row (n0+l), cols k + 2*half + {0,1}. Clamp OOB rows.
  int   wr     = n0 + l;
  float wvalid = (active && wr < N) ? 1.0f : 0.0f;
  int   wrc    = (wr < N) ? wr : (N - 1);
  const float* wp = W + (size_t)wrc * K + half * 2;

  // LDS read base: row (m*16 + l), col (kk + 2*half)
  const float* als = a_lds + (size_t)l * KCP + half * 2;

  v8f acc[8];
  v8f zero = {};
#pragma unroll
  for (int m = 0; m < 8; ++m) acc[m] = zero;

  for (int k0 = 0; k0 < K; k0 += KC) {
    // ---- cooperative stage A[:, k0:k0+KC] into LDS (float4 loads) ----
    __syncthreads();                                  // protect prior reads
#pragma unroll
    for (int i = tid; i < 128 * (KC / 4); i += 256) {
      int row = i >> 4;                               // KC/4 = 16 float4/row
      int c4  = i & 15;
      float4 v = *(const float4*)(A + (size_t)row * K + k0 + c4 * 4);
      float* dst = a_lds + row * KCP + c4 * 4;
      dst[0] = v.x; dst[1] = v.y; dst[2] = v.z; dst[3] = v.w;
    }
    __syncthreads();

    // ---- consume chunk: groups of 8 k-steps, B preloaded per group ----
#pragma unroll
    for (int g = 0; g < KC; g += 32) {                // 8 k-steps per group
      v2f bf[8];
#pragma unroll
      for (int j = 0; j < 8; ++j) {                   // batched -> load clause
        const float* p = wp + k0 + g + j * 4;
        bf[j].x = p[0];
        bf[j].y = p[1];
      }
#pragma unroll
      for (int j = 0; j < 8; ++j) {
        v2f bfr;
        bfr.x = wvalid * bf[j].x;
        bfr.y = wvalid * bf[j].y;
        v2f af[8];
#pragma unroll
        for (int m = 0; m < 8; ++m) {                 // batched -> ds clause
          const float* p = als + (size_t)(m * 16) * KCP + g + j * 4;
          af[m].x = p[0];
          af[m].y = p[1];
        }
#pragma unroll
        for (int m = 0; m < 8; ++m) {
          acc[m] = __builtin_amdgcn_wmma_f32_16x16x4_f32(
              false, af[m], false, bfr, (short)0, acc[m], false, false);
        }
      }
    }
  }

  const int col = n0 + l;
  if (active && col < N) {
    const float bval = bias ? bias[col] : 0.0f;
#pragma unroll
    for (int m = 0; m < 8; ++m) {
#pragma unroll
      for (int i = 0; i < 8; ++i) {
        int row = m * 16 + half * 8 + i;              // C layout per ISA 7.12.2
        float v = acc[m][i] + bval;
        if (fuse_tanh) v = tanhf(v);
        C[(size_t)row * ldc + col] = v;
      }
    }
  }
}

// ---------------------------------------------------------------------------
// GRU gates (PyTorch order r,z,n):  h' = (1-z)*n + z*h
// ---------------------------------------------------------------------------
__global__ void gru_gates_kernel(const float* __restrict__ gi,
                                 const float* __restrict__ gh,
                                 const float* __restrict__ h_prev,
                                 float* __restrict__ h_new,
                                 float* __restrict__ h_out) {
  int idx = blockIdx.x * blockDim.x + threadIdx.x;    // 128*512 threads
  int b = idx >> 9;
  int j = idx & 511;
  const float* gib = gi + (size_t)b * 3 * HID;
  const float* ghb = gh + (size_t)b * 3 * HID;
  float r = 1.0f / (1.0f + expf(-(gib[j]           + ghb[j])));
  float z = 1.0f / (1.0f + expf(-(gib[HID + j]     + ghb[HID + j])));
  float n = tanhf(gib[2 * HID + j] + r * ghb[2 * HID + j]);
  float h = (1.0f - z) * n + z * h_prev[idx];
  h_new[idx] = h;
  h_out[idx] = h;
}

// ---------------------------------------------------------------------------
// scores[s,b] = dot(encoder_outputs[b,s,:], q[b,:])   one wave per (b,s)
// ---------------------------------------------------------------------------
__global__ void attn_scores_kernel(const float* __restrict__ enc,
                                   const float* __restrict__ q,
                                   float* __restrict__ scores) {
  int wid  = (blockIdx.x * blockDim.x + threadIdx.x) >> 5;
  int lane = threadIdx.x & 31;
  if (wid >= BSZ * SRC_L) return;
  int b = wid >> 7;
  int s = wid & 127;
  const float* e  = enc + ((size_t)b * SRC_L + s) * HID;
  const float* qq = q + (size_t)b * HID;
  float sum = 0.0f;
  for (int h = lane; h < HID; h += 32) sum += e[h] * qq[h];
#pragma unroll
  for (int off = 16; off > 0; off >>= 1) sum += __shfl_xor(sum, off, 32);
  if (lane == 0) scores[s * BSZ + b] = sum;
}

// ---------------------------------------------------------------------------
// masked softmax over S (axis 0) for each batch column b. 128 threads/block.
// ---------------------------------------------------------------------------
__global__ void attn_softmax_kernel(const float* __restrict__ scores,
                                    const int* __restrict__ source_lens,
                                    float* __restrict__ a_out) {
  __shared__ float red[SRC_L];
  int b = blockIdx.x;
  int s = threadIdx.x;
  float v = scores[s * BSZ + b];
  v = (s < source_lens[b]) ? v : 0.0f;   // masked = mask * scores
  v = (v == 0.0f) ? -1e10f : v;          // where(masked == 0, -1e10, masked)
  red[s] = v;
  __syncthreads();
  for (int off = 64; off > 0; off >>= 1) {
    if (s < off) red[s] = fmaxf(red[s], red[s + off]);
    __syncthreads();
  }
  float mx = red[0];
  __syncthreads();
  float e = expf(v - mx);
  red[s] = e;
  __syncthreads();
  for (int off = 64; off > 0; off >>= 1) {
    if (s < off) red[s] += red[s + off];
    __syncthreads();
  }
  a_out[s * BSZ + b] = e / red[0];
}

// ---------------------------------------------------------------------------
// ctx_raw[b,h] = sum_s a[s,b] * enc[b,s,h].  One block per b, 512 threads.
// ---------------------------------------------------------------------------
__global__ void attn_ctx_kernel(const float* __restrict__ a,
                                const float* __restrict__ enc,
                                float* __restrict__ ctx) {
  __shared__ float as[SRC_L];
  int b = blockIdx.x;
  int h = threadIdx.x;                 // 0..511
  if (h < SRC_L) as[h] = a[h * BSZ + b];
  __syncthreads();
  const float* e = enc + (size_t)b * SRC_L * HID + h;
  float sum = 0.0f;
  for (int s = 0; s < SRC_L; ++s) sum += as[s] * e[(size_t)s * HID];
  ctx[b * HID + h] = sum;
}

// cat = [ctx_raw | h_new]  -> [128, 1024]
__global__ void build_cat_kernel(const float* __restrict__ ctx,
                                 const float* __restrict__ h,
                                 float* __restrict__ cat) {
  int idx = blockIdx.x * blockDim.x + threadIdx.x;    // 128*1024
  int b = idx >> 10;
  int j = idx & 1023;
  cat[idx] = (j < HID) ? ctx[b * HID + j] : h[b * HID + (j - HID)];
}

// ---------------------------------------------------------------------------
// In-place log-softmax over VOCAB per row. One block (256 thr) per row.
// ---------------------------------------------------------------------------
__global__ void log_softmax_kernel(float* __restrict__ logits) {
  __shared__ float red[256];
  int b = blockIdx.x;
  float* row = logits + (size_t)b * VOCAB;
  int t = threadIdx.x;
  float mx = -INFINITY;
  for (int j = t; j < VOCAB; j += 256) mx = fmaxf(mx, row[j]);
  red[t] = mx;
  __syncthreads();
  for (int off = 128; off > 0; off >>= 1) {
    if (t < off) red[t] = fmaxf(red[t], red[t + off]);
    __syncthreads();
  }
  mx = red[0];
  __syncthreads();
  float sum = 0.0f;
  for (int j = t; j < VOCAB; j += 256) sum += expf(row[j] - mx);
  red[t] = sum;
  __syncthreads();
  for (int off = 128; off > 0; off >>= 1) {
    if (t < off) red[t] += red[t + off];
    __syncthreads();
  }
  float lse = mx + logf(red[0]);
  for (int j = t; j < VOCAB; j += 256) row[j] = row[j] - lse;
}

// ---------------------------------------------------------------------------
extern "C" void kernel_launch(void* const* d_in, const int* in_sizes, int n_in,
                              void* d_out, int out_size, void* d_ws, size_t ws_size,
                              hipStream_t stream) {
  const int*   x        = (const int*)d_in[0];
  const float* dec_h    = (const float*)d_in[1];   // [1,128,512]
  const int*   src_lens = (const int*)d_in[2];
  const float* enc      = (const float*)d_in[3];   // [128,128,512]
  const float* ctx_vec  = (const float*)d_in[4];   // [128,512]
  const float* emb      = (const float*)d_in[5];   // [50257,512]
  const float* w_ih     = (const float*)d_in[6];   // [1536,1024]
  const float* w_hh     = (const float*)d_in[7];   // [1536,512]
  const float* b_ih     = (const float*)d_in[8];
  const float* b_hh     = (const float*)d_in[9];
  const float* Wa       = (const float*)d_in[10];  // [512,512]
  const float* Wc       = (const float*)d_in[11];  // [512,1024]
  const float* fc1_w    = (const float*)d_in[12];  // [50257,512]
  const float* fc1_b    = (const float*)d_in[13];

  float* out_logp   = (float*)d_out;                       // [128,50257]
  float* out_hidden = out_logp + (size_t)BSZ * VOCAB;      // [1,128,512]
  float* out_a      = out_hidden + BSZ * HID;              // [128,128] (S,B)
  float* out_ctx    = out_a + SRC_L * BSZ;                 // [128,512]

  float* ws      = (float*)d_ws;
  float* rnn_in  = ws;                                     // 128*1024
  float* gi      = rnn_in + BSZ * (EMBED + HID);           // 128*1536
  float* gh      = gi + BSZ * 3 * HID;                     // 128*1536
  float* h_new   = gh + BSZ * 3 * HID;                     // 128*512
  float* q       = h_new + BSZ * HID;                      // 128*512
  float* scores  = q + BSZ * HID;                          // 128*128
  float* ctx_raw = scores + SRC_L * BSZ;                   // 128*512
  float* cat     = ctx_raw + BSZ * HID;                    // 128*1024

  dim3 gblk(32, 8);  // 8 waves per block, one 16-wide N tile each

  build_rnn_in_kernel<<<BSZ * 1024 / 256, 256, 0, stream>>>(x, emb, ctx_vec, rnn_in);

  // gi = rnn_in @ w_ih^T + b_ih  (N=1536, K=1024); gh = h @ w_hh^T + b_hh (K=512)
  gemm128_wmma<<<dim3(12), gblk, 0, stream>>>(rnn_in, w_ih, b_ih, gi, 3 * HID, EMBED + HID, 3 * HID, 0);
  gemm128_wmma<<<dim3(12), gblk, 0, stream>>>(dec_h,  w_hh, b_hh, gh, 3 * HID, HID,        3 * HID, 0);

  gru_gates_kernel<<<BSZ * HID / 256, 256, 0, stream>>>(gi, gh, dec_h, h_new, out_hidden);

  // q = h' @ Wa^T   (N=512, K=512)
  gemm128_wmma<<<dim3(4), gblk, 0, stream>>>(h_new, Wa, nullptr, q, HID, HID, HID, 0);

  attn_scores_kernel<<<(BSZ * SRC_L * 32) / 256, 256, 0, stream>>>(enc, q, scores);
  attn_softmax_kernel<<<BSZ, SRC_L, 0, stream>>>(scores, src_lens, out_a);
  attn_ctx_kernel<<<BSZ, HID, 0, stream>>>(out_a, enc, ctx_raw);

  build_cat_kernel<<<BSZ * 1024 / 256, 256, 0, stream>>>(ctx_raw, h_new, cat);

  // ctx = tanh(cat @ Wc^T)  (N=512, K=1024), tanh fused into GEMM epilogue
  gemm128_wmma<<<dim3(4), gblk, 0, stream>>>(cat, Wc, nullptr, out_ctx, HID, 2 * HID, HID, 1);

  // logits = ctx @ fc1_w^T + fc1_b  (N=50257, K=512) -> HBM-bound weight stream
  {
    int ntiles = (VOCAB + 15) / 16;                 // 3142
    gemm128_wmma<<<dim3((ntiles + 7) / 8), gblk, 0, stream>>>(
        out_ctx, fc1_w, fc1_b, out_logp, VOCAB, HID, VOCAB, 0);
  }
  log_softmax_kernel<<<BSZ, 256, 0, stream>>>(out_logp);
}